// Model_54966991454770
// MI455X (gfx1250) — compile-verified
//
#include <hip/hip_runtime.h>
#include <hip/hip_bf16.h>

// Problem constants (from reference)
#define NNODES   128000      // B * NG
#define NGRAPH   64          // B
#define NPERG    2000        // NG
#define NEDGES   4096000     // B * EG
#define EPERG    64000       // EG
#define EMBD     64
#define G1D      32
#define G2D      32
#define KTOP     1600
#define DENSED   64
#define NCLSD    10
#define ECHUNK   2000        // edges staged per async chunk

typedef __attribute__((ext_vector_type(2))) float v2f;
typedef __attribute__((ext_vector_type(8))) float v8f;
typedef int v4i __attribute__((vector_size(16)));   // matches builtin's expected param type

#if defined(__has_builtin)
#if __has_builtin(__builtin_amdgcn_global_load_async_to_lds_b128)
#define USE_ASYNC_LDS 1
#endif
#endif

// ---------------------------------------------------------------- degree
__global__ __launch_bounds__(256) void deg_init_k(int* __restrict__ deg) {
    int i = blockIdx.x * 256 + threadIdx.x;
    if (i < NNODES) deg[i] = 1;   // self-loop
}

__global__ __launch_bounds__(256) void deg_count_k(const int* __restrict__ dst,
                                                   int* __restrict__ deg) {
    int e = blockIdx.x * 256 + threadIdx.x;
    if (e < NEDGES) atomicAdd(&deg[dst[e]], 1);
}

__global__ __launch_bounds__(256) void dinv_k(const int* __restrict__ deg,
                                              float* __restrict__ dinv) {
    int i = blockIdx.x * 256 + threadIdx.x;
    if (i < NNODES) {
        int d = deg[i];
        dinv[i] = (d > 0) ? rsqrtf((float)d) : 0.0f;
    }
}

// ---------------------------------------------------------------- embedding gather
__global__ __launch_bounds__(256) void gather_k(const int* __restrict__ x,
                                                const float* __restrict__ emb,
                                                float* __restrict__ h0) {
    int i = blockIdx.x * 256 + threadIdx.x;       // over NNODES*16
    if (i < NNODES * 16) {
        int node = i >> 4;
        int q    = i & 15;
        const float4* srcp = (const float4*)(emb + (long)x[node] * EMBD);
        float4* dstp       = (float4*)(h0 + (long)node * EMBD);
        dstp[q] = srcp[q];
    }
}

// ---------------------------------------------------------------- WMMA GEMM
// C[nrows x 32] = A[nrows x KDIM] @ W[KDIM x 32] via v_wmma_f32_16x16x4_f32.
// All B fragments are hoisted into registers once per wave (W is tiny), then
// each wave sweeps 4 row-tiles of 16 rows reusing them -> no per-step LDS reads.
template<int KDIM>
__global__ __launch_bounds__(256) void gemm_wmma_k(const float* __restrict__ A,
                                                   const float* __restrict__ W,
                                                   float* __restrict__ C,
                                                   int nrows) {
    __shared__ float w_s[KDIM * 32];
    for (int i = threadIdx.x; i < KDIM * 32; i += 256) w_s[i] = W[i];
    __syncthreads();

    int lane = threadIdx.x & 31;
    int wave = threadIdx.x >> 5;                  // 0..7
    int m  = lane & 15;                           // M index within half
    int kp = lane >> 4;                           // K-pair selector: 0 -> K{0,1}, 1 -> K{2,3}

    constexpr int KS = KDIM / 4;                  // k-steps
    v2f bf0[KS], bf1[KS];                         // B fragments for cols [0,16) and [16,32)
#pragma unroll
    for (int ks = 0; ks < KS; ++ks) {
        int k0 = ks * 4 + 2 * kp;
        bf0[ks].x = w_s[(k0    ) * 32 + m];
        bf0[ks].y = w_s[(k0 + 1) * 32 + m];
        bf1[ks].x = w_s[(k0    ) * 32 + 16 + m];
        bf1[ks].y = w_s[(k0 + 1) * 32 + 16 + m];
    }

    long wtile = ((long)blockIdx.x * 8 + wave) * 4;   // first 16-row tile of this wave
#pragma unroll
    for (int tI = 0; tI < 4; ++tI) {
        long row0 = (wtile + tI) * 16;
        if (row0 >= nrows) break;                 // wave-uniform
        const float* arow = A + (row0 + m) * KDIM + 2 * kp;
        v8f acc0 = {};
        v8f acc1 = {};
#pragma unroll
        for (int ks = 0; ks < KS; ++ks) {
            v2f a;
            a.x = arow[ks * 4 + 0];
            a.y = arow[ks * 4 + 1];
            acc0 = __builtin_amdgcn_wmma_f32_16x16x4_f32(false, a, false, bf0[ks],
                                                         (short)0, acc0, false, false);
            acc1 = __builtin_amdgcn_wmma_f32_16x16x4_f32(false, a, false, bf1[ks],
                                                         (short)0, acc1, false, false);
        }
        // C/D layout: VGPR r, lanes 0-15: M=r, N=lane; lanes 16-31: M=r+8, N=lane-16
#pragma unroll
        for (int r = 0; r < 8; ++r) {
            long row = row0 + r + 8 * kp;
            C[row * 32 + m]      = acc0[r];
            C[row * 32 + 16 + m] = acc1[r];
        }
    }
}

// ---------------------------------------------------------------- GCN aggregation
// One workgroup per graph. 2000x32 f32 accumulator (256 KB) lives in CDNA5's
// 320 KB LDS; edges scatter-add with ds_add_f32 atomics. Edge chunks are staged
// into LDS with gfx1250 async loads when available. Self-loop folded into the
// accumulator init. Output gets +bias and ReLU.
__global__ __launch_bounds__(1024) void aggregate_k(const float* __restrict__ t,
                                                    const float* __restrict__ dinv,
                                                    const int* __restrict__ src,
                                                    const int* __restrict__ dst,
                                                    const float* __restrict__ bias,
                                                    float* __restrict__ out) {
    __shared__ float acc[NPERG * 32];     // 256000 bytes
    __shared__ float dinv_s[NPERG];       //   8000 bytes
#if USE_ASYNC_LDS
    __shared__ int es_s[ECHUNK];          //   8000 bytes
    __shared__ int ed_s[ECHUNK];          //   8000 bytes
#endif

    int g    = blockIdx.x;
    int base = g * NPERG;

    for (int i = threadIdx.x; i < NPERG; i += 1024) dinv_s[i] = dinv[base + i];
    __syncthreads();

    // init with self-loop contribution: acc[n] = t[n] * dinv[n]^2
    for (int i = threadIdx.x; i < NPERG * 32; i += 1024) {
        int n = i >> 5, c = i & 31;
        float w = dinv_s[n] * dinv_s[n];
        acc[i] = t[(long)(base + n) * 32 + c] * w;
    }
    __syncthreads();

    long ebase = (long)g * EPERG;

#if USE_ASYNC_LDS
    for (int c0 = 0; c0 < EPERG; c0 += ECHUNK) {
        int tid = threadIdx.x;
        if (tid < 2 * (ECHUNK / 4)) {                    // 1000 lanes issue one b128 each
            int which = (tid >= (ECHUNK / 4));           // 0 = src, 1 = dst
            int idx   = which ? (tid - (ECHUNK / 4)) : tid;
            const int* gp = (which ? dst : src) + ebase + c0 + idx * 4;
            int*       lp = (which ? ed_s : es_s) + idx * 4;
            __builtin_amdgcn_global_load_async_to_lds_b128((v4i*)(void*)gp,
                                                           (v4i*)(void*)lp, 0, 0);
        }
        asm volatile("s_wait_asynccnt 0x0" ::: "memory");
        __syncthreads();

        for (int e = threadIdx.x; e < ECHUNK; e += 1024) {
            int s  = es_s[e];
            int d  = ed_s[e];
            int ls = s - base;
            int ld = d - base;
            float w = dinv_s[ls] * dinv_s[ld];
            const float4* tp = (const float4*)(t + (long)s * 32);
            float* ap = acc + ld * 32;
#pragma unroll
            for (int q = 0; q < 8; ++q) {
                float4 v = tp[q];
                atomicAdd(ap + q * 4 + 0, v.x * w);
                atomicAdd(ap + q * 4 + 1, v.y * w);
                atomicAdd(ap + q * 4 + 2, v.z * w);
                atomicAdd(ap + q * 4 + 3, v.w * w);
            }
        }
        __syncthreads();
    }
#else
    for (int e = threadIdx.x; e < EPERG; e += 1024) {
        int s  = src[ebase + e];
        int d  = dst[ebase + e];
        int ls = s - base;
        int ld = d - base;
        float w = dinv_s[ls] * dinv_s[ld];
        const float4* tp = (const float4*)(t + (long)s * 32);
        float* ap = acc + ld * 32;
#pragma unroll
        for (int q = 0; q < 8; ++q) {
            float4 v = tp[q];
            atomicAdd(ap + q * 4 + 0, v.x * w);
            atomicAdd(ap + q * 4 + 1, v.y * w);
            atomicAdd(ap + q * 4 + 2, v.z * w);
            atomicAdd(ap + q * 4 + 3, v.w * w);
        }
    }
    __syncthreads();
#endif

    for (int i = threadIdx.x; i < NPERG * 32; i += 1024) {
        float v = acc[i] + bias[i & 31];
        out[(long)base * 32 + i] = fmaxf(v, 0.0f);
    }
}

// ---------------------------------------------------------------- pool norm
__global__ void pool_norm_k(const float* __restrict__ pw, float* __restrict__ inv_norm) {
    __shared__ float s[32];
    int t = threadIdx.x;
    float v = pw[t];
    s[t] = v * v;
    __syncthreads();
    if (t == 0) {
        float sum = 0.0f;
        for (int i = 0; i < 32; ++i) sum += s[i];
        inv_norm[0] = rsqrtf(sum);
    }
}

// ---------------------------------------------------------------- score
__global__ __launch_bounds__(256) void score_k(const float* __restrict__ h2,
                                               const float* __restrict__ pw,
                                               const float* __restrict__ inv_norm,
                                               float* __restrict__ score) {
    int i = blockIdx.x * 256 + threadIdx.x;
    if (i < NNODES) {
        float s = 0.0f;
        const float* hp = h2 + (long)i * 32;
#pragma unroll
        for (int c = 0; c < 32; ++c) s += hp[c] * pw[c];
        score[i] = tanhf(s * inv_norm[0]);
    }
}

// ---------------------------------------------------------------- top-k + max pool
// Per graph: 32-bit radix select of the 1600-th largest score key, then
// column-wise max of h2[n,:]*score[n] over nodes with key >= threshold.
__global__ __launch_bounds__(256) void topk_max_k(const float* __restrict__ score,
                                                  const float* __restrict__ h2,
                                                  float* __restrict__ gout) {
    __shared__ unsigned keys[NPERG];
    __shared__ int cnt_s;
    __shared__ float partial[8 * 32];

    int g    = blockIdx.x;
    int tid  = threadIdx.x;
    int base = g * NPERG;

    for (int i = tid; i < NPERG; i += 256) {
        unsigned u = __float_as_uint(score[base + i]);
        keys[i] = (u & 0x80000000u) ? ~u : (u | 0x80000000u);   // order-preserving map
    }
    __syncthreads();

    unsigned prefix = 0;
    int need = KTOP;
    for (int bit = 31; bit >= 0; --bit) {
        if (tid == 0) cnt_s = 0;
        __syncthreads();
        unsigned bitv = 1u << bit;
        int local = 0;
        for (int i = tid; i < NPERG; i += 256) {
            unsigned k = keys[i];
            if ((((unsigned long long)(k ^ prefix)) >> (bit + 1)) == 0ull && (k & bitv))
                local++;
        }
        atomicAdd(&cnt_s, local);
        __syncthreads();
        int c = cnt_s;
        if (c >= need) prefix |= bitv; else need -= c;
        __syncthreads();
    }

    int col = tid & 31, grp = tid >> 5;
    float pm = -3.0e38f;
    for (int i = grp; i < NPERG; i += 8) {
        if (keys[i] >= prefix) {
            float sc = score[base + i];
            float v  = h2[(long)(base + i) * 32 + col] * sc;
            pm = fmaxf(pm, v);
        }
    }
    partial[grp * 32 + col] = pm;
    __syncthreads();
    if (tid < 32) {
        float m = partial[col];
#pragma unroll
        for (int r = 1; r < 8; ++r) m = fmaxf(m, partial[r * 32 + col]);
        gout[g * 32 + col] = m;
    }
}

// ---------------------------------------------------------------- MLP head
__global__ __launch_bounds__(256) void head_k(const float* __restrict__ g,
                                              const float* __restrict__ dW,
                                              const float* __restrict__ db,
                                              const float* __restrict__ oW,
                                              const float* __restrict__ ob,
                                              float* __restrict__ out) {
    __shared__ float d_s[NGRAPH * DENSED];
    for (int i = threadIdx.x; i < NGRAPH * DENSED; i += 256) {
        int b = i >> 6, j = i & 63;
        float s = db[j];
#pragma unroll
        for (int k = 0; k < G2D; ++k) s += g[b * G2D + k] * dW[k * DENSED + j];
        d_s[i] = fmaxf(s, 0.0f);
    }
    __syncthreads();
    for (int i = threadIdx.x; i < NGRAPH * NCLSD; i += 256) {
        int b = i / NCLSD, c = i % NCLSD;
        float s = ob[c];
#pragma unroll
        for (int j = 0; j < DENSED; ++j) s += d_s[b * DENSED + j] * oW[j * NCLSD + c];
        out[i] = s;
    }
}

// ---------------------------------------------------------------- launch
extern "C" void kernel_launch(void* const* d_in, const int* in_sizes, int n_in,
                              void* d_out, int out_size, void* d_ws, size_t ws_size,
                              hipStream_t stream) {
    const int*   x      = (const int*)d_in[0];
    const int*   edges  = (const int*)d_in[1];          // [2, E]
    // d_in[2] = batch (unused; layout is implicit)
    const float* emb    = (const float*)d_in[3];
    const float* W1     = (const float*)d_in[4];
    const float* b1     = (const float*)d_in[5];
    const float* W2     = (const float*)d_in[6];
    const float* b2     = (const float*)d_in[7];
    const float* pool_w = (const float*)d_in[8];
    const float* denseW = (const float*)d_in[9];
    const float* denseb = (const float*)d_in[10];
    const float* outW   = (const float*)d_in[11];
    const float* outb   = (const float*)d_in[12];
    float* outp = (float*)d_out;

    const int* srcp = edges;
    const int* dstp = edges + NEDGES;

    char* w = (char*)d_ws;
    float* h0   = (float*)w;  w += (size_t)NNODES * EMBD * 4;
    float* t1   = (float*)w;  w += (size_t)NNODES * G1D * 4;
    float* h1   = (float*)w;  w += (size_t)NNODES * G1D * 4;
    float* t2   = (float*)w;  w += (size_t)NNODES * G2D * 4;
    float* h2   = (float*)w;  w += (size_t)NNODES * G2D * 4;
    int*   deg  = (int*)w;    w += (size_t)NNODES * 4;
    float* dinv = (float*)w;  w += (size_t)NNODES * 4;
    float* scr  = (float*)w;  w += (size_t)NNODES * 4;
    float* gbuf = (float*)w;  w += (size_t)NGRAPH * G2D * 4;
    float* invn = (float*)w;  w += 256;

    // degrees / normalization
    deg_init_k<<<(NNODES + 255) / 256, 256, 0, stream>>>(deg);
    deg_count_k<<<(NEDGES + 255) / 256, 256, 0, stream>>>(dstp, deg);
    dinv_k<<<(NNODES + 255) / 256, 256, 0, stream>>>(deg, dinv);

    // embedding gather
    gather_k<<<(NNODES * 16 + 255) / 256, 256, 0, stream>>>(x, emb, h0);

    // layer 1: t1 = h0 @ W1 ; h1 = relu(agg(t1) + b1)
    gemm_wmma_k<EMBD><<<NNODES / (16 * 4 * 8), 256, 0, stream>>>(h0, W1, t1, NNODES);
    aggregate_k<<<NGRAPH, 1024, 0, stream>>>(t1, dinv, srcp, dstp, b1, h1);

    // layer 2: t2 = h1 @ W2 ; h2 = relu(agg(t2) + b2)
    gemm_wmma_k<G1D><<<NNODES / (16 * 4 * 8), 256, 0, stream>>>(h1, W2, t2, NNODES);
    aggregate_k<<<NGRAPH, 1024, 0, stream>>>(t2, dinv, srcp, dstp, b2, h2);

    // scores
    pool_norm_k<<<1, 32, 0, stream>>>(pool_w, invn);
    score_k<<<(NNODES + 255) / 256, 256, 0, stream>>>(h2, pool_w, invn, scr);

    // top-k + weighted max pool per graph
    topk_max_k<<<NGRAPH, 256, 0, stream>>>(scr, h2, gbuf);

    // MLP head
    head_k<<<1, 256, 0, stream>>>(gbuf, denseW, denseb, outW, outb, outp);
}